// EncoderLayer_22703197127369
// MI455X (gfx1250) — compile-verified
//
#include <hip/hip_runtime.h>
#include <math.h>

typedef float v2f __attribute__((ext_vector_type(2)));
typedef float v8f __attribute__((ext_vector_type(8)));

// B=2, L=32, D=512, H=8, DK=64, DS=512, DE=1536, BL=64
#define LQ   32
#define DQ   512
#define DSQ  512
#define DEQ  1536
#define BLQ  64
#define LN_EPS 1e-5f

__device__ __forceinline__ float wave_sum(float v) {
#pragma unroll
  for (int off = 16; off > 0; off >>= 1) v += __shfl_xor(v, off, 32);
  return v;
}

// ---------------- LayerNorm: one wave per row (64 rows, D=512, 16 elems/lane)
__global__ void ln_kernel(const float* __restrict__ X, const float* __restrict__ g,
                          const float* __restrict__ b, float* __restrict__ O) {
  const int row = blockIdx.x;
  const int lane = threadIdx.x;
  float vals[16];
  float s = 0.f;
#pragma unroll
  for (int j = 0; j < 16; ++j) {
    vals[j] = X[row * DQ + lane + 32 * j];
    s += vals[j];
  }
  const float mean = wave_sum(s) * (1.0f / DQ);
  float q = 0.f;
#pragma unroll
  for (int j = 0; j < 16; ++j) { float d = vals[j] - mean; q += d * d; }
  const float var = wave_sum(q) * (1.0f / DQ);
  const float rs = rsqrtf(var + LN_EPS);
#pragma unroll
  for (int j = 0; j < 16; ++j) {
    const int c = lane + 32 * j;
    O[row * DQ + c] = (vals[j] - mean) * rs * g[c] + b[c];
  }
}

// ---------------- fp32 WMMA GEMM: C[64,N] = A[64,K] @ W[K,N] + bias (+resid)
// One wave computes a 16x64 tile via 4 accumulators; K stepped by 4 using
// V_WMMA_F32_16X16X4_F32. grid = (N/64, 4), block = 32 (one wave32).
__global__ void gemm_wmma_f32(const float* __restrict__ A, int lda,
                              const float* __restrict__ W, int ldw,
                              const float* __restrict__ bias,
                              const float* __restrict__ resid, int ldr,
                              float* __restrict__ C, int ldc, int K) {
  const int lane  = threadIdx.x;
  const int half  = lane >> 4;   // 0 or 1 -> K offset 0/2 within the 4-step
  const int idx16 = lane & 15;   // M row (A) / N col (B,C,D)
  const int m0 = blockIdx.y * 16;
  const int n0 = blockIdx.x * 64;

  v8f acc[4];
#pragma unroll
  for (int t = 0; t < 4; ++t) acc[t] = (v8f){0.f,0.f,0.f,0.f,0.f,0.f,0.f,0.f};

  const float* arow = A + (size_t)(m0 + idx16) * lda + 2 * half;
  for (int kb = 0; kb < K; kb += 4) {
    v2f a = *(const v2f*)(arow + kb);
    const float* wp = W + (size_t)(kb + 2 * half) * ldw + n0 + idx16;
    v2f b0, b1, b2, b3;
    b0.x = wp[0];  b0.y = wp[ldw];
    b1.x = wp[16]; b1.y = wp[ldw + 16];
    b2.x = wp[32]; b2.y = wp[ldw + 32];
    b3.x = wp[48]; b3.y = wp[ldw + 48];
    acc[0] = __builtin_amdgcn_wmma_f32_16x16x4_f32(false, a, false, b0, (short)0, acc[0], false, false);
    acc[1] = __builtin_amdgcn_wmma_f32_16x16x4_f32(false, a, false, b1, (short)0, acc[1], false, false);
    acc[2] = __builtin_amdgcn_wmma_f32_16x16x4_f32(false, a, false, b2, (short)0, acc[2], false, false);
    acc[3] = __builtin_amdgcn_wmma_f32_16x16x4_f32(false, a, false, b3, (short)0, acc[3], false, false);
  }

#pragma unroll
  for (int r = 0; r < 8; ++r) {
    const int m = m0 + r + 8 * half;
#pragma unroll
    for (int t = 0; t < 4; ++t) {
      const int n = n0 + t * 16 + idx16;
      float v = acc[t][r] + bias[n];
      if (resid) v += resid[(size_t)m * ldr + n];
      C[(size_t)m * ldc + n] = v;
    }
  }
}

// ---------------- RoPE in-place on q and k ((BL,D) layout, head-major cols)
__global__ void rope_kernel(float* __restrict__ qb, float* __restrict__ kb) {
  const int idx = blockIdx.x * blockDim.x + threadIdx.x; // 0 .. 2*64*256-1
  float* t = (idx >= BLQ * 256) ? kb : qb;
  const int r = idx & (BLQ * 256 - 1);
  const int bl = r >> 8;          // row
  const int p  = r & 255;         // pair index within row
  const int h  = p >> 5;
  const int j  = p & 31;          // rotary pair
  const int base = bl * DQ + h * 64 + j;
  const float v1 = t[base], v2 = t[base + 32];
  const float pos  = (float)(bl & 31);
  const float invf = __expf(-0.28782288f * (float)j); // 10000^(-j/32)
  const float f = pos * invf;
  const float cs = __cosf(f), sn = __sinf(f);
  t[base]      = v1 * cs - v2 * sn;
  t[base + 32] = v2 * cs + v1 * sn;
}

// ---------------- attention: one wave per (b,h); lane = query index
__global__ void attn_kernel(const float* __restrict__ Q, const float* __restrict__ K,
                            const float* __restrict__ V, const int* __restrict__ mask,
                            float* __restrict__ ctx) {
  __shared__ float qs[32][65];
  __shared__ float ps[32][33];
  const int lane = threadIdx.x;
  const int b = blockIdx.x >> 3, h = blockIdx.x & 7;
  const int row = b * 32 + lane;
  const float* Qb = Q + (size_t)row * DQ + h * 64;
  for (int d = 0; d < 64; ++d) qs[lane][d] = Qb[d];
  for (int kk = 0; kk < 32; ++kk) {
    const float* Kb = K + (size_t)(b * 32 + kk) * DQ + h * 64;
    float s = 0.f;
    for (int d = 0; d < 64; ++d) s += qs[lane][d] * Kb[d];
    s *= 0.125f; // 1/sqrt(64)
    if (mask[b * 1024 + lane * 32 + kk] == 0) s = -1e9f;
    ps[lane][kk] = s;
  }
  float mx = -3.0e38f;
  for (int kk = 0; kk < 32; ++kk) mx = fmaxf(mx, ps[lane][kk]);
  float sum = 0.f;
  for (int kk = 0; kk < 32; ++kk) {
    float p = __expf(ps[lane][kk] - mx);
    ps[lane][kk] = p; sum += p;
  }
  const float inv = 1.0f / sum;
  for (int d = 0; d < 64; ++d) {
    float acc = 0.f;
    for (int kk = 0; kk < 32; ++kk)
      acc += ps[lane][kk] * V[(size_t)(b * 32 + kk) * DQ + h * 64 + d];
    ctx[(size_t)row * DQ + h * 64 + d] = acc * inv;
  }
}

// ---------------- silu in place on gate half of tmp (cols 0..1535 of 3072)
__global__ void silu_kernel(float* __restrict__ tmp) {
  const int idx = blockIdx.x * blockDim.x + threadIdx.x; // 64*1536
  const int bl = idx / DEQ, e = idx - bl * DEQ;
  float* p = tmp + (size_t)bl * (2 * DEQ) + e;
  const float g = *p;
  *p = g * (1.0f / (1.0f + __expf(-g)));
}

// ---------------- delta column: wave dot per row + softplus
__global__ void delta_kernel(const float* __restrict__ xssm /*lda=3072*/,
                             const float* __restrict__ ssm_w,
                             const float* __restrict__ ssm_b,
                             float* __restrict__ dlt) {
  const int bl = blockIdx.x, lane = threadIdx.x;
  float s = 0.f;
  for (int k = lane; k < DEQ; k += 32)
    s += xssm[(size_t)bl * (2 * DEQ) + k] * ssm_w[(size_t)k * 2049 + 2048];
  s = wave_sum(s);
  if (lane == 0) {
    const float d = s + ssm_b[2048];
    dlt[bl] = (d > 20.0f) ? d : log1pf(__expf(d)); // softplus
  }
}

// ---------------- complex SSM scan: one wave per (b,e); 16 states/lane
__global__ void scan_kernel(const float* __restrict__ Are_g, const float* __restrict__ Aim_g,
                            const float* __restrict__ ssmo /* (BL,1024): C_re|C_im */,
                            const float* __restrict__ dlt,
                            const float* __restrict__ xssm /* lda=3072 */,
                            float* __restrict__ y /* (BL,DE) */) {
  const int be = blockIdx.x;
  const int b = be / DEQ, e = be - b * DEQ;
  const int lane = threadIdx.x;
  float Are[16], Aim[16], hre[16], him[16];
#pragma unroll
  for (int j = 0; j < 16; ++j) {
    const int s = lane + 32 * j;
    Are[j] = Are_g[(size_t)e * DSQ + s];
    Aim[j] = Aim_g[(size_t)e * DSQ + s];
    hre[j] = 0.f; him[j] = 0.f;
  }
#pragma unroll 1
  for (int l = 0; l < LQ; ++l) {
    const int bl = b * LQ + l;
    const float d  = dlt[bl];
    const float xt = xssm[(size_t)bl * (2 * DEQ) + e];
    float part = 0.f;
#pragma unroll
    for (int j = 0; j < 16; ++j) {
      const float dre = d * Are[j], dim = d * Aim[j];
      const float er = __expf(dre);
      const float cs = __cosf(dim), sn = __sinf(dim);
      const float abre = er * cs, abim = er * sn;          // A_bar = exp(dA)
      const float mag = sqrtf(dre * dre + dim * dim);
      // small-|dA| branch: d*(1 + dA/2)
      const float bres = d * (1.0f + 0.5f * dre);
      const float bims = d * 0.5f * dim;
      // large branch: d*(A_bar-1)/(dA+eps)
      const float denr = dre + 1.1920929e-7f, deni = dim;
      const float inv = 1.0f / (denr * denr + deni * deni);
      const float nre = d * (abre - 1.0f), nim = d * abim;
      const float breb = (nre * denr + nim * deni) * inv;
      const float bimb = (nim * denr - nre * deni) * inv;
      const float Bre = (mag < 1e-3f) ? bres : breb;
      const float Bim = (mag < 1e-3f) ? bims : bimb;
      // h = A_bar*h + B_bar*x
      const float nhre = abre * hre[j] - abim * him[j] + Bre * xt;
      const float nhim = abre * him[j] + abim * hre[j] + Bim * xt;
      hre[j] = nhre; him[j] = nhim;
      // y += Re(h * conj(C)) = h_re*C_re + h_im*C_im
      const int s = lane + 32 * j;
      const float Cre = ssmo[(size_t)bl * 1024 + s];
      const float Cim = ssmo[(size_t)bl * 1024 + 512 + s];
      part += nhre * Cre + nhim * Cim;
    }
    part = wave_sum(part);
    if (lane == 0) y[(size_t)bl * DEQ + e] = part;
  }
}

// ---------------- z = (y + x_ssm * D) * gate, in place on y
__global__ void zmul_kernel(float* __restrict__ y, const float* __restrict__ tmp,
                            const float* __restrict__ Dp) {
  const int idx = blockIdx.x * blockDim.x + threadIdx.x; // 64*1536
  const int bl = idx / DEQ, e = idx - bl * DEQ;
  const float xs = tmp[(size_t)bl * (2 * DEQ) + DEQ + e];
  const float g  = tmp[(size_t)bl * (2 * DEQ) + e];
  y[idx] = (y[idx] + xs * Dp[e]) * g;
}

extern "C" void kernel_launch(void* const* d_in, const int* in_sizes, int n_in,
                              void* d_out, int out_size, void* d_ws, size_t ws_size,
                              hipStream_t stream) {
  const float* x      = (const float*)d_in[0];
  const int*   mask   = (const int*)  d_in[1];
  const float* ln1_g  = (const float*)d_in[2];
  const float* ln1_b  = (const float*)d_in[3];
  const float* ln2_g  = (const float*)d_in[4];
  const float* ln2_b  = (const float*)d_in[5];
  const float* wq     = (const float*)d_in[6];
  const float* bq     = (const float*)d_in[7];
  const float* wk     = (const float*)d_in[8];
  const float* bk     = (const float*)d_in[9];
  const float* wv     = (const float*)d_in[10];
  const float* bv     = (const float*)d_in[11];
  const float* wo     = (const float*)d_in[12];
  const float* bo     = (const float*)d_in[13];
  const float* in_w   = (const float*)d_in[14];
  const float* in_b   = (const float*)d_in[15];
  const float* out_w  = (const float*)d_in[16];
  const float* out_b  = (const float*)d_in[17];
  const float* A_re   = (const float*)d_in[18];
  const float* A_im   = (const float*)d_in[19];
  const float* ssm_w  = (const float*)d_in[20];
  const float* ssm_b  = (const float*)d_in[21];
  const float* Dp     = (const float*)d_in[22];
  float* out = (float*)d_out;

  float* ws   = (float*)d_ws;
  float* xn1  = ws;            // 64*512
  float* qb   = ws + 32768;
  float* kb   = ws + 65536;
  float* vb   = ws + 98304;
  float* ctx  = ws + 131072;
  float* x1   = ws + 163840;
  float* xn2  = ws + 196608;
  float* tmp  = ws + 229376;   // 64*3072 (gate in cols 0..1535, x_ssm 1536..)
  float* ssmo = ws + 425984;   // 64*1024 (C_re | C_im)
  float* dlt  = ws + 491520;   // 64
  float* yb   = ws + 491584;   // 64*1536

  // --- attention sublayer ---
  ln_kernel<<<BLQ, 32, 0, stream>>>(x, ln1_g, ln1_b, xn1);
  gemm_wmma_f32<<<dim3(8, 4), 32, 0, stream>>>(xn1, DQ, wq, DQ, bq, nullptr, 0, qb, DQ, DQ);
  gemm_wmma_f32<<<dim3(8, 4), 32, 0, stream>>>(xn1, DQ, wk, DQ, bk, nullptr, 0, kb, DQ, DQ);
  gemm_wmma_f32<<<dim3(8, 4), 32, 0, stream>>>(xn1, DQ, wv, DQ, bv, nullptr, 0, vb, DQ, DQ);
  rope_kernel<<<128, 256, 0, stream>>>(qb, kb);
  attn_kernel<<<16, 32, 0, stream>>>(qb, kb, vb, mask, ctx);
  gemm_wmma_f32<<<dim3(8, 4), 32, 0, stream>>>(ctx, DQ, wo, DQ, bo, x, DQ, x1, DQ, DQ);

  // --- LRNN/SSM sublayer ---
  ln_kernel<<<BLQ, 32, 0, stream>>>(x1, ln2_g, ln2_b, xn2);
  gemm_wmma_f32<<<dim3(48, 4), 32, 0, stream>>>(xn2, DQ, in_w, 2 * DEQ, in_b, nullptr, 0,
                                                tmp, 2 * DEQ, DQ);
  silu_kernel<<<(BLQ * DEQ) / 256, 256, 0, stream>>>(tmp);
  // SSM projection: only C_re|C_im columns (1024..2047) are consumed by the scan
  gemm_wmma_f32<<<dim3(16, 4), 32, 0, stream>>>(tmp + DEQ, 2 * DEQ, ssm_w + 2 * DSQ, 4 * DSQ + 1,
                                                ssm_b + 2 * DSQ, nullptr, 0, ssmo, 1024, DEQ);
  delta_kernel<<<BLQ, 32, 0, stream>>>(tmp + DEQ, ssm_w, ssm_b, dlt);
  scan_kernel<<<2 * DEQ, 32, 0, stream>>>(A_re, A_im, ssmo, dlt, tmp + DEQ, yb);
  zmul_kernel<<<(BLQ * DEQ) / 256, 256, 0, stream>>>(yb, tmp, Dp);
  gemm_wmma_f32<<<dim3(8, 4), 32, 0, stream>>>(yb, DEQ, out_w, DQ, out_b, x1, DQ, out, DQ, DEQ);
}